// NODEDynamics_50036368998573
// MI455X (gfx1250) — compile-verified
//
#include <hip/hip_runtime.h>

// ---------------------------------------------------------------------------
// GCN (3x GCNConv + 2x BatchNorm/ReLU) for MI455X / gfx1250.
// f32 end-to-end; GEMMs via V_WMMA_F32_16X16X4_F32 (wave32 matrix cores).
// ---------------------------------------------------------------------------

typedef __attribute__((ext_vector_type(2))) float v2f;
typedef __attribute__((ext_vector_type(8))) float v8f;

#define FDIM 64
#define BN_EPS 1e-5f

// ---------------------------------------------------------------- utilities
__device__ __forceinline__ void atomic_add_f32(float* p, float v) {
  __hip_atomic_fetch_add(p, v, __ATOMIC_RELAXED, __HIP_MEMORY_SCOPE_AGENT);
}

// -------------------------------------------------------- degree / gcn_norm
__global__ void init_deg_kernel(float* __restrict__ deg, int n) {
  int i = blockIdx.x * blockDim.x + threadIdx.x;
  if (i < n) deg[i] = 1.0f;  // self-loop weight 1.0
}

__global__ void deg_accum_kernel(const long long* __restrict__ ei,
                                 const float* __restrict__ w,
                                 float* __restrict__ deg, int E) {
  int e = blockIdx.x * blockDim.x + threadIdx.x;
  if (e < E) {
    int col = (int)ei[(long)E + e];
    atomic_add_f32(&deg[col], w[e]);
  }
}

__global__ void finalize_dinv_kernel(float* __restrict__ deg, int n) {
  int i = blockIdx.x * blockDim.x + threadIdx.x;
  if (i < n) {
    float d = deg[i];
    deg[i] = (d > 0.0f) ? rsqrtf(d) : 0.0f;
  }
}

__global__ void norm_kernel(const long long* __restrict__ ei,
                            const float* __restrict__ w,
                            const float* __restrict__ dinv,
                            float* __restrict__ norm, int E) {
  int e = blockIdx.x * blockDim.x + threadIdx.x;
  if (e < E) {
    int r = (int)ei[e];
    int c = (int)ei[(long)E + e];
    norm[e] = dinv[r] * w[e] * dinv[c];
  }
}

// -------------------------------------------------------------- WMMA GEMM
// Y[N,64] = X[N,64] @ W[64,64], f32.  One wave handles a 16-row strip and all
// four 16-column tiles (64 v_wmma_f32_16x16x4_f32 per wave; A reused 4x).
//
// A (16x4 f32) layout: lanes 0-15 rows M=0..15; VGPR0/1 hold K={0,1} for the
// low half-wave and K={2,3} for the high half-wave.
// B (4x16 f32) layout: mirrored (lanes give N, half-wave selects K pair).
// C/D (16x16 f32): lane = N (mod 16), VGPR v = M = v + 8*(lane/16).
__global__ __launch_bounds__(256) void gemm64_wmma_kernel(
    const float* __restrict__ X, const float* __restrict__ W,
    float* __restrict__ Y, int nRowTiles) {
  const int lane = threadIdx.x & 31;
  const int wv   = threadIdx.x >> 5;
  const int rt   = blockIdx.x * 8 + wv;
  if (rt >= nRowTiles) return;  // uniform per-wave exit: EXEC stays all-ones

  const int half = lane >> 4;   // 0 | 1
  const int l    = lane & 15;
  const long rowBase = (long)(rt * 16 + l) * FDIM;

  v8f acc0 = {}, acc1 = {}, acc2 = {}, acc3 = {};

#pragma unroll
  for (int k = 0; k < FDIM; k += 4) {
    const int kb = k + half * 2;
    v2f av;
    av[0] = X[rowBase + kb];
    av[1] = X[rowBase + kb + 1];

    const float* __restrict__ w0 = W + (long)kb * FDIM;
    const float* __restrict__ w1 = w0 + FDIM;

    v2f bv;
    bv[0] = w0[l];      bv[1] = w1[l];
    acc0 = __builtin_amdgcn_wmma_f32_16x16x4_f32(false, av, false, bv,
                                                 (short)0, acc0, false, false);
    bv[0] = w0[16 + l]; bv[1] = w1[16 + l];
    acc1 = __builtin_amdgcn_wmma_f32_16x16x4_f32(false, av, false, bv,
                                                 (short)0, acc1, false, false);
    bv[0] = w0[32 + l]; bv[1] = w1[32 + l];
    acc2 = __builtin_amdgcn_wmma_f32_16x16x4_f32(false, av, false, bv,
                                                 (short)0, acc2, false, false);
    bv[0] = w0[48 + l]; bv[1] = w1[48 + l];
    acc3 = __builtin_amdgcn_wmma_f32_16x16x4_f32(false, av, false, bv,
                                                 (short)0, acc3, false, false);
  }

#pragma unroll
  for (int v = 0; v < 8; ++v) {
    const int m = v + half * 8;
    float* __restrict__ yr = Y + (long)(rt * 16 + m) * FDIM + l;
    yr[0]  = acc0[v];
    yr[16] = acc1[v];
    yr[32] = acc2[v];
    yr[48] = acc3[v];
  }
}

// Scalar tail for row counts not divisible by 16 (not used when N%16==0).
__global__ void gemm64_tail_kernel(const float* __restrict__ X,
                                   const float* __restrict__ W,
                                   float* __restrict__ Y, int rowStart, int nRows) {
  long gid = blockIdx.x * (long)blockDim.x + threadIdx.x;
  long total = (long)(nRows - rowStart) * FDIM;
  if (gid >= total) return;
  int r = rowStart + (int)(gid >> 6);
  int n = (int)(gid & 63);
  float s = 0.0f;
  for (int k = 0; k < FDIM; ++k) s += X[(long)r * FDIM + k] * W[(long)k * FDIM + n];
  Y[(long)r * FDIM + n] = s;
}

// -------------------------------------------------------------- aggregation
// agg[i,f] = bias[f] + dinv[i]^2 * xw[i,f]   (self-loop term, no atomics)
__global__ void agg_init_kernel(const float* __restrict__ xw,
                                const float* __restrict__ dinv,
                                const float* __restrict__ bias,
                                float* __restrict__ agg, int n) {
  long idx = blockIdx.x * (long)blockDim.x + threadIdx.x;
  if (idx >= (long)n * FDIM) return;
  int i = (int)(idx >> 6);
  int f = (int)(idx & 63);
  float di = dinv[i];
  agg[idx] = bias[f] + di * di * xw[idx];
}

// agg[col,f] += norm[e] * xw[row,f] ; 64 consecutive lanes per edge so the
// gather and the f32 atomics are both fully coalesced (L2-resident).
__global__ void agg_edges_kernel(const long long* __restrict__ ei,
                                 const float* __restrict__ norm,
                                 const float* __restrict__ xw,
                                 float* __restrict__ agg, int E) {
  long gid = blockIdx.x * (long)blockDim.x + threadIdx.x;
  if (gid >= (long)E * FDIM) return;
  int e = (int)(gid >> 6);
  int f = (int)(gid & 63);
  int row = (int)ei[e];
  int col = (int)ei[(long)E + e];
  float v = norm[e] * xw[(long)row * FDIM + f];
  atomic_add_f32(&agg[(long)col * FDIM + f], v);
}

// -------------------------------------------------------------- batch norm
__global__ void zero_stats_kernel(float* __restrict__ stats) {
  if (threadIdx.x < 128) stats[threadIdx.x] = 0.0f;
}

__global__ __launch_bounds__(256) void bn_stats_kernel(
    const float* __restrict__ agg, float* __restrict__ stats, int n) {
  __shared__ float ssum[256];
  __shared__ float ssq[256];
  const int f   = threadIdx.x & 63;
  const int grp = threadIdx.x >> 6;  // 0..3 row-groups per block
  float s = 0.0f, q = 0.0f;
  for (int r = blockIdx.x * 4 + grp; r < n; r += gridDim.x * 4) {
    float v = agg[(long)r * FDIM + f];
    s += v;
    q += v * v;
  }
  ssum[threadIdx.x] = s;
  ssq[threadIdx.x]  = q;
  __syncthreads();
  if (threadIdx.x < 64) {
    float S = ssum[f] + ssum[f + 64] + ssum[f + 128] + ssum[f + 192];
    float Q = ssq[f] + ssq[f + 64] + ssq[f + 128] + ssq[f + 192];
    atomic_add_f32(&stats[f], S);
    atomic_add_f32(&stats[64 + f], Q);
  }
}

__global__ void bn_apply_relu_kernel(const float* __restrict__ agg,
                                     const float* __restrict__ gamma,
                                     const float* __restrict__ beta,
                                     const float* __restrict__ stats,
                                     float* __restrict__ out, int n) {
  long idx = blockIdx.x * (long)blockDim.x + threadIdx.x;
  if (idx >= (long)n * FDIM) return;
  int f = (int)(idx & 63);
  float invN = 1.0f / (float)n;
  float mean = stats[f] * invN;
  float var  = stats[64 + f] * invN - mean * mean;  // biased, matches jnp.var
  float sc   = gamma[f] * rsqrtf(var + BN_EPS);
  float v    = (agg[idx] - mean) * sc + beta[f];
  out[idx] = v > 0.0f ? v : 0.0f;
}

// -------------------------------------------------------------- output pack
// Tuple (dz[N,64], edge_index[2,E], edge_weight[E], batch_vector[N]) flattened.
__global__ void write_out_kernel(const float* __restrict__ dz,
                                 const long long* __restrict__ ei,
                                 const float* __restrict__ w,
                                 const int* __restrict__ batch,
                                 float* __restrict__ out,
                                 int N, int E, long out_size) {
  long i = blockIdx.x * (long)blockDim.x + threadIdx.x;
  long nd = (long)N * FDIM;
  if (i < nd) {
    if (i < out_size) out[i] = dz[i];
    return;
  }
  long j = i - nd;
  if (j < 2L * E) {
    long o = nd + j;
    if (o < out_size) out[o] = (float)ei[j];
    return;
  }
  j -= 2L * E;
  if (j < E) {
    long o = nd + 2L * E + j;
    if (o < out_size) out[o] = w[j];
    return;
  }
  j -= E;
  if (j < N) {
    long o = nd + 3L * E + j;
    if (o < out_size) out[o] = (float)batch[j];
  }
}

// ---------------------------------------------------------------- launcher
extern "C" void kernel_launch(void* const* d_in, const int* in_sizes, int n_in,
                              void* d_out, int out_size, void* d_ws, size_t ws_size,
                              hipStream_t stream) {
  // inputs (setup_inputs order)
  const float*     x0    = (const float*)d_in[1];       // node_features [N,64]
  const long long* ei    = (const long long*)d_in[2];   // edge_index [2,E] int64
  const float*     ew    = (const float*)d_in[3];       // edge_weight [E]
  const int*       batch = (const int*)d_in[4];         // batch_vector [N]
  const float*     W0    = (const float*)d_in[5];
  const float*     b0    = (const float*)d_in[6];
  const float*     g0    = (const float*)d_in[7];
  const float*     be0   = (const float*)d_in[8];
  const float*     W1    = (const float*)d_in[9];
  const float*     b1    = (const float*)d_in[10];
  const float*     g1    = (const float*)d_in[11];
  const float*     be1   = (const float*)d_in[12];
  const float*     Wf    = (const float*)d_in[13];
  const float*     bf    = (const float*)d_in[14];

  const int N = in_sizes[1] / FDIM;
  const int E = in_sizes[2] / 2;

  // workspace layout (floats)
  float* ws    = (float*)d_ws;
  float* dinv  = ws;                 ws += N;
  float* norm  = ws;                 ws += E;
  float* stats = ws;                 ws += 128;
  float* xw    = ws;                 ws += (long)N * FDIM;
  float* agg   = ws;                 ws += (long)N * FDIM;
  float* h     = ws;

  const int T = 256;
  auto cdiv = [](long a, long b) { return (int)((a + b - 1) / b); };

  // gcn_norm
  init_deg_kernel<<<cdiv(N, T), T, 0, stream>>>(dinv, N);
  deg_accum_kernel<<<cdiv(E, T), T, 0, stream>>>(ei, ew, dinv, E);
  finalize_dinv_kernel<<<cdiv(N, T), T, 0, stream>>>(dinv, N);
  norm_kernel<<<cdiv(E, T), T, 0, stream>>>(ei, ew, dinv, norm, E);

  const float* Win[3]  = {W0, W1, Wf};
  const float* bin[3]  = {b0, b1, bf};
  const float* gin[2]  = {g0, g1};
  const float* bein[2] = {be0, be1};

  const int rowTiles = N / 16;
  const int rem      = N % 16;
  const float* xin = x0;

  for (int layer = 0; layer < 3; ++layer) {
    if (rowTiles > 0)
      gemm64_wmma_kernel<<<cdiv(rowTiles, 8), 256, 0, stream>>>(xin, Win[layer], xw, rowTiles);
    if (rem > 0)
      gemm64_tail_kernel<<<cdiv((long)rem * FDIM, T), T, 0, stream>>>(xin, Win[layer], xw,
                                                                      rowTiles * 16, N);
    agg_init_kernel<<<cdiv((long)N * FDIM, T), T, 0, stream>>>(xw, dinv, bin[layer], agg, N);
    agg_edges_kernel<<<cdiv((long)E * FDIM, T), T, 0, stream>>>(ei, norm, xw, agg, E);

    if (layer < 2) {
      zero_stats_kernel<<<1, 128, 0, stream>>>(stats);
      bn_stats_kernel<<<512, 256, 0, stream>>>(agg, stats, N);
      bn_apply_relu_kernel<<<cdiv((long)N * FDIM, T), T, 0, stream>>>(agg, gin[layer],
                                                                      bein[layer], stats, h, N);
      xin = h;
    }
  }

  long totalOut = (long)N * FDIM + 3L * E + N;
  write_out_kernel<<<cdiv(totalOut, T), T, 0, stream>>>(agg, ei, ew, batch, (float*)d_out,
                                                        N, E, (long)out_size);
}